// PGA_60438779790045
// MI455X (gfx1250) — compile-verified
//
#include <hip/hip_runtime.h>
#include <math.h>

typedef __attribute__((ext_vector_type(2))) float v2f;
typedef __attribute__((ext_vector_type(8))) float v8f;

// Problem constants (from reference): B=8, BS=2048, N=16, M=128, L=32
#define CB   8
#define CBS  2048
#define CN   16
#define CM   128
#define CL   32

// LDS layout (floats)
#define OFF_WA   0      // 128*32  = 4096
#define OFF_WD   4096   // 8*32*16 = 4096
#define OFF_HW   8192   // 8*16*32 = 4096
#define OFF_A    12288  // 8*16*16 = 2048
#define OFF_SCR  14336  // 8*16*32 = 4096 (augmented Gauss-Jordan / logdet scratch)
#define OFF_T    18432  // 8*16*16 = 2048
#define OFF_U    20480  // 8*16*32 = 4096
#define OFF_F2   24576  // 128*32  = 4096
#define OFF_RED  28672  // 16
#define OFF_LAM  28688  // 16
#define OFF_PRM  28704  // 16 (ints)
#define LDS_FLOATS 28736

// ---------------------------------------------------------------------------
// One 16x16 output tile of C += op(A) * op(B), fp32 WMMA 16x16x4, wave32.
// A-operand layout (32-bit A 16x4): lanes 0-15 -> K=k,k+1 ; lanes 16-31 -> K=k+2,k+3
// B-operand (4x16): lane holds column n = lane&15, 2 K values per the half.
// C/D: lane col = lane&15, VGPR r holds row r (lanes<16) / r+8 (lanes>=16).
// ---------------------------------------------------------------------------
__device__ __forceinline__ v8f wmma_tile(const float* A, int lda, bool ta,
                                         const float* Bp, int ldb, bool tb,
                                         int K, v8f acc, int lane)
{
    const int half = lane >> 4;      // 0 or 1
    const int idx  = lane & 15;
    for (int k = 0; k < K; k += 4) {
        const int ka = k + half * 2;
        v2f a, b;
        if (!ta) { a.x = A[idx * lda + ka];      a.y = A[idx * lda + ka + 1]; }
        else     { a.x = A[ka * lda + idx];      a.y = A[(ka + 1) * lda + idx]; }
        if (!tb) { b.x = Bp[ka * ldb + idx];     b.y = Bp[(ka + 1) * ldb + idx]; }
        else     { b.x = Bp[idx * ldb + ka];     b.y = Bp[idx * ldb + ka + 1]; }
        acc = __builtin_amdgcn_wmma_f32_16x16x4_f32(false, a, false, b,
                                                    (short)0, acc, false, false);
    }
    return acc;
}

__device__ __forceinline__ void store_tile(float* C, int ldc, v8f c, int lane, float scale)
{
    const int col = lane & 15;
    const int mb  = (lane >> 4) * 8;
    #pragma unroll
    for (int r = 0; r < 8; ++r) C[(mb + r) * ldc + col] = c[r] * scale;
}

// In-wave Gauss-Jordan inverse of SPD 16x16 (augmented [G|I], row stride 32).
// lane == column (32 lanes cover 32 cols). Wave32 lockstep + volatile => in-order LDS.
__device__ __forceinline__ void gj_invert16(float* Wp, int lane)
{
    volatile float* W = Wp;
    for (int k = 0; k < 16; ++k) {
        float inv = 1.0f / W[k * 32 + k];
        W[k * 32 + lane] *= inv;
        for (int r = 0; r < 16; ++r) {
            if (r == k) continue;
            float f = W[r * 32 + k];
            W[r * 32 + lane] -= f * W[k * 32 + lane];
        }
    }
}

// In-wave LU (no pivoting, SPD, diag>=1) log-determinant; 16x16 row stride 32.
__device__ __forceinline__ float lu_logdet16(float* Wp, int lane)
{
    volatile float* W = Wp;
    float ldet = 0.f;
    for (int k = 0; k < 16; ++k) {
        float piv = W[k * 32 + k];
        ldet += __logf(fmaxf(piv, 1e-30f));
        float inv = 1.0f / piv;
        if (lane < 16) {
            for (int r = k + 1; r < 16; ++r) {
                float f = W[r * 32 + k] * inv;
                W[r * 32 + lane] -= f * W[k * 32 + lane];
            }
        }
    }
    return ldet;
}

// p_b = ||WA(128x32) @ WD_b(32x16)||_F^2 accumulated straight out of the WMMA
// accumulators (no LDS round trip); wave wv handles batch b = wv.
__device__ __forceinline__ void power_scale_partial(const float* WA, const float* WD,
                                                    float* RED, int wv, int lane)
{
    float p = 0.f;
    for (int mt = 0; mt < 8; ++mt) {
        v8f acc = {0.f,0.f,0.f,0.f,0.f,0.f,0.f,0.f};
        acc = wmma_tile(WA + mt * 16 * CL, CL, false, WD + wv * 512, CN, false, CL, acc, lane);
        #pragma unroll
        for (int r = 0; r < 8; ++r) p += acc[r] * acc[r];
    }
    for (int o = 16; o; o >>= 1) p += __shfl_xor(p, o, 32);
    if (lane == 0) RED[wv] = p;
    __syncthreads();
}

__global__ void __launch_bounds__(256)
pga_kernel(const float* __restrict__ h, const float* __restrict__ hyp,
           const float* __restrict__ wd_init, const int* __restrict__ nit_p,
           float* __restrict__ out)
{
    extern __shared__ float sm[];
    const int s    = blockIdx.x;          // sample in [0, 2048)
    const int tid  = threadIdx.x;
    const int lane = tid & 31;            // wave32
    const int wv   = tid >> 5;            // wave id == batch b owned by this wave
    const int NIT  = nit_p[0];

    float* WA  = sm + OFF_WA;
    float* WD  = sm + OFF_WD;
    float* HW  = sm + OFF_HW;
    float* AA  = sm + OFF_A;
    float* SCR = sm + OFF_SCR;
    float* TT  = sm + OFF_T;
    float* UU  = sm + OFF_U;
    float* F2  = sm + OFF_F2;
    float* RED = sm + OFF_RED;
    float* LAM = sm + OFF_LAM;
    int*   PRM = (int*)(sm + OFF_PRM);

    const size_t hs = (size_t)CN * CM;    // 2048 floats per (b,s) slice of h
    const float* hB[CB];
    #pragma unroll
    for (int b = 0; b < CB; ++b) hB[b] = h + ((size_t)b * CBS + s) * hs;

    // Warm L2 for this sample's h slices (192MB L2 holds the whole h).
    #pragma unroll
    for (int b = 0; b < CB; ++b) {
        const char* p = (const char*)hB[b];
        for (int off = tid * 64; off < (int)(hs * 4); off += 256 * 64)
            __builtin_prefetch(p + off, 0, 1);
    }

    // ============================ SETUP =====================================
    // mean over batches -> MH (16x128), aliased onto F2
    float* MH = F2;
    for (int e = tid; e < CN * CM; e += 256) {
        float acc = 0.f;
        #pragma unroll
        for (int b = 0; b < CB; ++b) acc += hB[b][e];
        MH[e] = acc * 0.125f;
    }
    __syncthreads();

    // Gram matrix G0 = MH*MH^T (wave0, WMMA), then cyclic Jacobi eigensolve.
    float* G0 = TT;         // 16x16
    float* U0 = TT + 256;   // 16x16 eigenvectors (columns)
    if (wv == 0) {
        v8f acc = {0.f,0.f,0.f,0.f,0.f,0.f,0.f,0.f};
        acc = wmma_tile(MH, CM, false, MH, CM, true, CM, acc, lane);
        store_tile(G0, 16, acc, lane, 1.f);
        volatile float* G = G0;
        volatile float* U = U0;
        if (lane < 16)
            for (int j = 0; j < 16; ++j) U[lane * 16 + j] = (lane == j) ? 1.f : 0.f;
        for (int sweep = 0; sweep < 12; ++sweep) {
            for (int p = 0; p < 15; ++p)
            for (int q = p + 1; q < 16; ++q) {
                float apq = G[p * 16 + q];
                if (fabsf(apq) < 1e-10f) continue;           // wave-uniform
                float app = G[p * 16 + p], aqq = G[q * 16 + q];
                float th = 0.5f * (aqq - app) / apq;
                float t  = ((th >= 0.f) ? 1.f : -1.f) / (fabsf(th) + sqrtf(1.f + th * th));
                float c  = rsqrtf(1.f + t * t);
                float sn = t * c;
                if (lane < 16) {                              // rotate rows p,q
                    int j = lane;
                    float gpj = G[p * 16 + j], gqj = G[q * 16 + j];
                    G[p * 16 + j] = c * gpj - sn * gqj;
                    G[q * 16 + j] = sn * gpj + c * gqj;
                }
                if (lane < 16) {                              // rotate cols p,q (+ eigvecs)
                    int j = lane;
                    float gjp = G[j * 16 + p], gjq = G[j * 16 + q];
                    G[j * 16 + p] = c * gjp - sn * gjq;
                    G[j * 16 + q] = sn * gjp + c * gjq;
                    float ujp = U[j * 16 + p], ujq = U[j * 16 + q];
                    U[j * 16 + p] = c * ujp - sn * ujq;
                    U[j * 16 + q] = sn * ujp + c * ujq;
                }
            }
        }
        if (lane < 16) {                                      // sort eigenpairs desc
            float ml = G[lane * 16 + lane];
            int rank = 0;
            for (int j = 0; j < 16; ++j) {
                float lj = G[j * 16 + j];
                if (lj > ml || (lj == ml && j < lane)) rank++;
            }
            PRM[rank] = lane;
            LAM[rank] = ml;
        }
    }
    __syncthreads();

    // Right singular vectors v_i = MH^T u_i / sigma_i  (16 x 128), aliased onto UU.
    float* V = UU;
    for (int e = tid; e < 16 * CM; e += 256) {
        int i = e >> 7, c = e & 127;
        float lam = LAM[i];
        float rs  = (lam > 1e-12f) ? rsqrtf(lam) : 0.f;
        int   ui  = PRM[i];
        float acc = 0.f;
        #pragma unroll
        for (int j = 0; j < 16; ++j) acc += MH[j * CM + c] * U0[j * 16 + ui];
        V[e] = acc * rs;
    }
    __syncthreads();

    // wa[r][c] = vh[r][c]: rows < 16 from computed V; rows >= 16 deterministic
    // orthogonal completion (e_r projected off span(V), normalized).
    for (int e = tid; e < CM * CL; e += 256) {
        int r = e >> 5, c = e & 31;
        float val;
        if (r < 16) val = V[r * CM + c];
        else {
            float acc = (r == c) ? 1.f : 0.f;
            float ss  = 0.f;
            #pragma unroll
            for (int i = 0; i < 16; ++i) {
                float vir = V[i * CM + r];
                acc -= vir * V[i * CM + c];
                ss  += vir * vir;
            }
            val = acc * rsqrtf(fmaxf(1.f - ss, 1e-12f));
        }
        WA[e] = val;
    }
    // stage wd_init
    for (int e = tid; e < CB * CL * CN; e += 256) {
        int b = e >> 9, rc = e & 511;
        WD[e] = wd_init[((size_t)b * CBS + s) * (CL * CN) + rc];
    }
    __syncthreads();

    // wd = power_scale(wa, wd_init) * wd_init ; scale = sqrt(n*b/p) = sqrt(128/p)
    power_scale_partial(WA, WD, RED, wv, lane);
    {
        float pt = 0.f;
        #pragma unroll
        for (int b = 0; b < CB; ++b) pt += RED[b];
        float sc = sqrtf(128.f / pt);
        for (int e = tid; e < CB * CL * CN; e += 256) WD[e] *= sc;
    }
    __syncthreads();

    // ========================== ITERATIONS ==================================
    const float* hg = hB[wv];
    for (int it = 0; it < NIT; ++it) {
        const float mu0 = hyp[it * (CB + 1)];
        const float mub = hyp[it * (CB + 1) + 1 + wv];

        // ---- Phase A (wave = batch): hw = h_b @ wa (16x32, K=128) ----
        for (int nt = 0; nt < 2; ++nt) {
            v8f acc = {0.f,0.f,0.f,0.f,0.f,0.f,0.f,0.f};
            acc = wmma_tile(hg, CM, false, WA + nt * 16, CL, false, CM, acc, lane);
            store_tile(HW + wv * 512 + nt * 16, CL, acc, lane, 1.f);
        }
        // A = hw @ wd (16x16, K=32)
        {
            v8f acc = {0.f,0.f,0.f,0.f,0.f,0.f,0.f,0.f};
            acc = wmma_tile(HW + wv * 512, CL, false, WD + wv * 512, CN, false, CL, acc, lane);
            store_tile(AA + wv * 256, CN, acc, lane, 1.f);
        }
        // G = I + A A^T -> augmented [G|I] in SCR ; invert (symmetric => GinvT == Ginv)
        {
            v8f acc = {0.f,0.f,0.f,0.f,0.f,0.f,0.f,0.f};
            acc = wmma_tile(AA + wv * 256, CN, false, AA + wv * 256, CN, true, CN, acc, lane);
            const int col = lane & 15, mb = (lane >> 4) * 8;
            float* W = SCR + wv * 512;
            #pragma unroll
            for (int r = 0; r < 8; ++r) {
                W[(mb + r) * 32 + col]      = acc[r] + ((mb + r) == col ? 1.f : 0.f);
                W[(mb + r) * 32 + 16 + col] = ((mb + r) == col) ? 1.f : 0.f;
            }
        }
        gj_invert16(SCR + wv * 512, lane);
        // T1 = Ginv @ A
        {
            v8f acc = {0.f,0.f,0.f,0.f,0.f,0.f,0.f,0.f};
            acc = wmma_tile(SCR + wv * 512 + 16, 32, false, AA + wv * 256, CN, false, CN, acc, lane);
            store_tile(TT + wv * 256, CN, acc, lane, 1.f);
        }
        // U = T1 @ wd^T (16x32)
        for (int nt = 0; nt < 2; ++nt) {
            v8f acc = {0.f,0.f,0.f,0.f,0.f,0.f,0.f,0.f};
            acc = wmma_tile(TT + wv * 256, CN, false, WD + wv * 512 + nt * 256, CN, true, CN, acc, lane);
            store_tile(UU + wv * 512 + nt * 16, CL, acc, lane, 1.f);
        }
        __syncthreads();

        // ---- Phase B (wave = M-row-tile): f2 = mean_b h_b^T @ U_b (128x32) ----
        for (int nt = 0; nt < 2; ++nt) {
            v8f acc = {0.f,0.f,0.f,0.f,0.f,0.f,0.f,0.f};
            #pragma unroll
            for (int b = 0; b < CB; ++b)
                acc = wmma_tile(hB[b] + wv * 16, CM, true, UU + b * 512 + nt * 16, CL, false, CN, acc, lane);
            store_tile(F2 + (wv * 16) * CL + nt * 16, CL, acc, lane, 0.125f);
        }
        __syncthreads();

        // ---- Phase C: wa_t = wa + mu0 * f2 (in place) ----
        for (int e = tid; e < CM * CL; e += 256) WA[e] += mu0 * F2[e];
        __syncthreads();

        // ---- Phase D: wa_new = power_scale * wa_t ----
        power_scale_partial(WA, WD, RED, wv, lane);
        {
            float pt = 0.f;
            #pragma unroll
            for (int b = 0; b < CB; ++b) pt += RED[b];
            float sc = sqrtf(128.f / pt);
            for (int e = tid; e < CM * CL; e += 256) WA[e] *= sc;
        }
        __syncthreads();

        // ---- Phase E (wave = batch): hw2, A2, G2inv, T2, gwd, wd_t ----
        for (int nt = 0; nt < 2; ++nt) {
            v8f acc = {0.f,0.f,0.f,0.f,0.f,0.f,0.f,0.f};
            acc = wmma_tile(hg, CM, false, WA + nt * 16, CL, false, CM, acc, lane);
            store_tile(HW + wv * 512 + nt * 16, CL, acc, lane, 1.f);
        }
        {
            v8f acc = {0.f,0.f,0.f,0.f,0.f,0.f,0.f,0.f};
            acc = wmma_tile(HW + wv * 512, CL, false, WD + wv * 512, CN, false, CL, acc, lane);
            store_tile(AA + wv * 256, CN, acc, lane, 1.f);
        }
        {
            v8f acc = {0.f,0.f,0.f,0.f,0.f,0.f,0.f,0.f};
            acc = wmma_tile(AA + wv * 256, CN, false, AA + wv * 256, CN, true, CN, acc, lane);
            const int col = lane & 15, mb = (lane >> 4) * 8;
            float* W = SCR + wv * 512;
            #pragma unroll
            for (int r = 0; r < 8; ++r) {
                W[(mb + r) * 32 + col]      = acc[r] + ((mb + r) == col ? 1.f : 0.f);
                W[(mb + r) * 32 + 16 + col] = ((mb + r) == col) ? 1.f : 0.f;
            }
        }
        gj_invert16(SCR + wv * 512, lane);
        {
            v8f acc = {0.f,0.f,0.f,0.f,0.f,0.f,0.f,0.f};
            acc = wmma_tile(SCR + wv * 512 + 16, 32, false, AA + wv * 256, CN, false, CN, acc, lane);
            store_tile(TT + wv * 256, CN, acc, lane, 1.f);
        }
        // gwd = hw2^T @ T2 / 8  (32x16); fold into wd_t = wd + mu_b * gwd (in place)
        for (int mt = 0; mt < 2; ++mt) {
            v8f acc = {0.f,0.f,0.f,0.f,0.f,0.f,0.f,0.f};
            acc = wmma_tile(HW + wv * 512 + mt * 16, CL, true, TT + wv * 256, CN, false, CN, acc, lane);
            const int col = lane & 15, mb = (lane >> 4) * 8;
            #pragma unroll
            for (int r = 0; r < 8; ++r)
                WD[wv * 512 + (mt * 16 + mb + r) * CN + col] += mub * 0.125f * acc[r];
        }
        __syncthreads();

        // ---- Phase F: wd_new = power_scale * wd_t ----
        power_scale_partial(WA, WD, RED, wv, lane);
        {
            float pt = 0.f;
            #pragma unroll
            for (int b = 0; b < CB; ++b) pt += RED[b];
            float sc = sqrtf(128.f / pt);
            for (int e = tid; e < CB * CL * CN; e += 256) WD[e] *= sc;
        }
        __syncthreads();

        // ---- Phase G: A3 = hw2 @ wd_new ; rate = sum_b logdet(I + A3 A3^T)/8 ----
        {
            v8f acc = {0.f,0.f,0.f,0.f,0.f,0.f,0.f,0.f};
            acc = wmma_tile(HW + wv * 512, CL, false, WD + wv * 512, CN, false, CL, acc, lane);
            store_tile(AA + wv * 256, CN, acc, lane, 1.f);
        }
        {
            v8f acc = {0.f,0.f,0.f,0.f,0.f,0.f,0.f,0.f};
            acc = wmma_tile(AA + wv * 256, CN, false, AA + wv * 256, CN, true, CN, acc, lane);
            const int col = lane & 15, mb = (lane >> 4) * 8;
            float* W = SCR + wv * 512;
            #pragma unroll
            for (int r = 0; r < 8; ++r)
                W[(mb + r) * 32 + col] = acc[r] + ((mb + r) == col ? 1.f : 0.f);
        }
        float ld = lu_logdet16(SCR + wv * 512, lane);
        if (lane == 0) RED[wv] = ld;
        __syncthreads();
        if (tid == 0) {
            float rt = 0.f;
            for (int b = 0; b < CB; ++b) rt += RED[b];
            out[(size_t)s * NIT + it] = rt * 0.125f;   // sum_rate[s][it]
        }
        __syncthreads();
    }

    // ====================== FINAL OUTPUT WRITES =============================
    const size_t owa = (size_t)CBS * NIT;
    const size_t owd = owa + (size_t)CB * CBS * CM * CL;
    #pragma unroll
    for (int b = 0; b < CB; ++b) {       // wa broadcast over batch dim
        float* dst = out + owa + ((size_t)b * CBS + s) * (CM * CL);
        for (int e = tid; e < CM * CL; e += 256) dst[e] = WA[e];
    }
    for (int e = tid; e < CB * CL * CN; e += 256) {
        int b = e >> 9, rc = e & 511;
        out[owd + ((size_t)b * CBS + s) * (CL * CN) + rc] = WD[b * 512 + rc];
    }
}

extern "C" void kernel_launch(void* const* d_in, const int* in_sizes, int n_in,
                              void* d_out, int out_size, void* d_ws, size_t ws_size,
                              hipStream_t stream)
{
    const float* h   = (const float*)d_in[0];
    const float* hyp = (const float*)d_in[1];
    const float* wdi = (const float*)d_in[2];
    const int*   nit = (const int*)d_in[3];
    float* out = (float*)d_out;
    (void)in_sizes; (void)n_in; (void)out_size; (void)d_ws; (void)ws_size;

    const size_t lds_bytes = (size_t)LDS_FLOATS * sizeof(float); // ~112 KB < 320 KB/WGP
    pga_kernel<<<dim3(CBS), dim3(256), lds_bytes, stream>>>(h, hyp, wdi, nit, out);
}